// GATNet_67697274520361
// MI455X (gfx1250) — compile-verified
//
#include <hip/hip_runtime.h>
#include <hip/hip_bf16.h>

// -------- GAT 2-layer for MI455X (gfx1250), fp32 WMMA path --------
// Precision note: V_WMMA_F32_16X16X4_F32 keeps the GEMMs in exact fp32,
// matching the fp32 JAX reference; the graph phase is atomic/bandwidth bound.

#define NEG_SLOPE 0.2f
#define F_IN   128
#define H1C1   160   // 8 heads * 20 ch
#define HEADS1 8
#define CH1    20
#define NCLS   16

typedef float v2f __attribute__((ext_vector_type(2)));
typedef float v8f __attribute__((ext_vector_type(8)));

// ---------------- helpers ----------------

__device__ __forceinline__ void edge_sd(const int* __restrict__ ei, int E, int idx,
                                        int& s, int& d) {
    if (idx < E) { s = ei[idx]; d = ei[E + idx]; }
    else         { s = idx - E; d = idx - E; }        // self loops appended
}

// order-preserving float<->uint encoding (for atomicMax on floats incl. negatives)
__device__ __forceinline__ unsigned f2ord(float f) {
    unsigned b = __float_as_uint(f);
    return (b & 0x80000000u) ? ~b : (b | 0x80000000u);
}
__device__ __forceinline__ float ord2f(unsigned u) {
    return (u & 0x80000000u) ? __uint_as_float(u ^ 0x80000000u)
                             : __uint_as_float(~u);
}

__device__ __forceinline__ float lrelu(float v) {
    return (v > 0.f) ? v : NEG_SLOPE * v;
}

// ---------------- kernels ----------------

// grid-stride zero fill of the accumulation region of the workspace
__global__ void k_zero(float* __restrict__ p, size_t n) {
    size_t i = (size_t)blockIdx.x * blockDim.x + threadIdx.x;
    size_t stride = (size_t)gridDim.x * blockDim.x;
    for (; i < n; i += stride) p[i] = 0.f;
}

// C[M x Nc] = A[M x K] * B[K x Nc], fp32 WMMA 16x16x4, one wave per 16x16 tile.
// M % 16 == 0, Nc % 16 == 0, K % 4 == 0 (true for all shapes here).
// A frag (ISA 7.12.2, 32-bit A 16x4): lanes 0-15 -> K=0,1 ; lanes 16-31 -> K=2,3 ; M = lane&15
// B frag mirrors A (rows striped across lanes): v0 = row k0+khalf, v1 = row k0+khalf+1, N = lane&15
// C/D: VGPR r -> M = r + (lane<16 ? 0 : 8), N = lane&15
__global__ void k_gemm_wmma_f32(const float* __restrict__ A,
                                const float* __restrict__ B,
                                float* __restrict__ C,
                                int M, int K, int Nc) {
    int wave = blockIdx.x * (blockDim.x >> 5) + (threadIdx.x >> 5);
    int ntN = Nc >> 4;
    int ntM = M >> 4;
    if (wave >= ntM * ntN) return;              // wave-uniform branch (EXEC stays all-1 for WMMA)
    int tm = wave / ntN, tn = wave % ntN;
    int m0 = tm << 4, n0 = tn << 4;

    int lane  = threadIdx.x & 31;
    int lr    = lane & 15;                      // M (for A/C) or N (for B/C) within tile
    int hi    = lane >> 4;                      // 0: lanes 0-15, 1: lanes 16-31
    int khalf = hi << 1;                        // 0 or 2

    const float* __restrict__ arow = A + (size_t)(m0 + lr) * K;
    v8f c = {};
    for (int k0 = 0; k0 < K; k0 += 4) {
        v2f a, b;
        a.x = arow[k0 + khalf];
        a.y = arow[k0 + khalf + 1];
        b.x = B[(size_t)(k0 + khalf)     * Nc + n0 + lr];
        b.y = B[(size_t)(k0 + khalf + 1) * Nc + n0 + lr];
        c = __builtin_amdgcn_wmma_f32_16x16x4_f32(
                /*neg_a=*/false, a, /*neg_b=*/false, b,
                /*c_mod=*/(short)0, c, /*reuse_a=*/false, /*reuse_b=*/false);
    }
#pragma unroll
    for (int r = 0; r < 8; ++r) {
        int row = m0 + r + (hi << 3);
        C[(size_t)row * Nc + n0 + lr] = c[r];
    }
}

// a_src[n,h] = sum_c h[n,h,c]*att_src[h,c] ; same for a_dst. one thread per (node,head)
__global__ void k_attn_proj(const float* __restrict__ h,
                            const float* __restrict__ att_s,
                            const float* __restrict__ att_d,
                            float* __restrict__ a_s, float* __restrict__ a_d,
                            int n, int heads, int ch) {
    int i = blockIdx.x * blockDim.x + threadIdx.x;
    if (i >= n * heads) return;
    int node = i / heads, hh = i - node * heads;
    const float* hp = h + (size_t)node * heads * ch + (size_t)hh * ch;
    const float* as = att_s + hh * ch;
    const float* ad = att_d + hh * ch;
    float ss = 0.f, sd = 0.f;
    for (int c = 0; c < ch; ++c) {
        float v = hp[c];
        ss += v * as[c];
        sd += v * ad[c];
    }
    a_s[i] = ss;
    a_d[i] = sd;
}

// pass A: per-destination running max of leaky-relu logits (ordered-uint atomicMax)
__global__ void k_edge_max(const int* __restrict__ ei, int E, int Etot, int heads,
                           const float* __restrict__ a_s, const float* __restrict__ a_d,
                           unsigned* __restrict__ menc) {
    int i = blockIdx.x * blockDim.x + threadIdx.x;
    if (i >= Etot * heads) return;
    int e = i / heads, hh = i - e * heads;
    int s, d; edge_sd(ei, E, e, s, d);
    float v = lrelu(a_s[s * heads + hh] + a_d[d * heads + hh]);
    atomicMax(&menc[(size_t)d * heads + hh], f2ord(v));
}

// pass B: denom[dst,h] += exp(e - m[dst,h])
__global__ void k_edge_expsum(const int* __restrict__ ei, int E, int Etot, int heads,
                              const float* __restrict__ a_s, const float* __restrict__ a_d,
                              const unsigned* __restrict__ menc,
                              float* __restrict__ denom) {
    int i = blockIdx.x * blockDim.x + threadIdx.x;
    if (i >= Etot * heads) return;
    int e = i / heads, hh = i - e * heads;
    int s, d; edge_sd(ei, E, e, s, d);
    float v = lrelu(a_s[s * heads + hh] + a_d[d * heads + hh]);
    float w = expf(v - ord2f(menc[(size_t)d * heads + hh]));
    atomicAdd(&denom[(size_t)d * heads + hh], w);
}

// pass C: agg[dst,h,:] += alpha * h[src,h,:]
__global__ void k_edge_scatter(const int* __restrict__ ei, int E, int Etot,
                               int heads, int ch,
                               const float* __restrict__ a_s, const float* __restrict__ a_d,
                               const unsigned* __restrict__ menc,
                               const float* __restrict__ denom,
                               const float* __restrict__ h,
                               float* __restrict__ agg) {
    int i = blockIdx.x * blockDim.x + threadIdx.x;
    if (i >= Etot * heads) return;
    int e = i / heads, hh = i - e * heads;
    int s, d; edge_sd(ei, E, e, s, d);
    float v = lrelu(a_s[s * heads + hh] + a_d[d * heads + hh]);
    float alpha = expf(v - ord2f(menc[(size_t)d * heads + hh]))
                  / denom[(size_t)d * heads + hh];
    const float* hs = h   + (size_t)s * heads * ch + (size_t)hh * ch;
    float*       ag = agg + (size_t)d * heads * ch + (size_t)hh * ch;
    for (int c = 0; c < ch; ++c)
        atomicAdd(&ag[c], alpha * hs[c]);
}

// in-place x = relu(x + b) over [n, hc]
__global__ void k_bias_relu(float* __restrict__ x, const float* __restrict__ b,
                            int n, int hc) {
    int i = blockIdx.x * blockDim.x + threadIdx.x;
    if (i >= n * hc) return;
    x[i] = fmaxf(x[i] + b[i % hc], 0.f);
}

// out = agg + b (final layer, no relu)
__global__ void k_bias_out(const float* __restrict__ agg, const float* __restrict__ b,
                           float* __restrict__ out, int n, int hc) {
    int i = blockIdx.x * blockDim.x + threadIdx.x;
    if (i >= n * hc) return;
    out[i] = agg[i] + b[i % hc];
}

// ---------------- launch ----------------

extern "C" void kernel_launch(void* const* d_in, const int* in_sizes, int n_in,
                              void* d_out, int out_size, void* d_ws, size_t ws_size,
                              hipStream_t stream) {
    const float* x        = (const float*)d_in[0];
    const int*   ei       = (const int*)  d_in[1];   // [2,E] int32 (src row, dst row)
    const float* W1       = (const float*)d_in[2];
    const float* att_src1 = (const float*)d_in[3];
    const float* att_dst1 = (const float*)d_in[4];
    const float* b1       = (const float*)d_in[5];
    const float* W2       = (const float*)d_in[6];
    const float* att_src2 = (const float*)d_in[7];
    const float* att_dst2 = (const float*)d_in[8];
    const float* b2       = (const float*)d_in[9];
    float* out = (float*)d_out;

    const int N    = in_sizes[0] / F_IN;  // 50000
    const int E    = in_sizes[1] / 2;     // 800000
    const int Etot = E + N;               // + self loops

    // ---- workspace carve-out (floats) ----
    float* ws  = (float*)d_ws;
    float* h1  = ws;                                   // [N,160]
    float* as1 = h1  + (size_t)N * H1C1;               // [N,8]
    float* ad1 = as1 + (size_t)N * HEADS1;             // [N,8]
    float* h2  = ad1 + (size_t)N * HEADS1;             // [N,16]
    float* as2 = h2  + (size_t)N * NCLS;               // [N]
    float* ad2 = as2 + (size_t)N;                      // [N]
    // zero-filled accumulation region starts here
    float*    zbase = ad2 + (size_t)N;
    float*    agg1  = zbase;                           // [N,160]
    float*    den1  = agg1 + (size_t)N * H1C1;         // [N,8]
    unsigned* m1    = (unsigned*)(den1 + (size_t)N * HEADS1); // [N,8]
    float*    agg2  = (float*)(m1 + (size_t)N * HEADS1);      // [N,16]
    float*    den2  = agg2 + (size_t)N * NCLS;         // [N]
    unsigned* m2    = (unsigned*)(den2 + (size_t)N);   // [N]
    size_t zcount = (size_t)N * (H1C1 + HEADS1 + HEADS1 + NCLS + 1 + 1); // 194*N

    const int T = 256;

    // 0) zero accumulators (m-enc init 0 == below f2ord(-inf); every node has a self loop)
    k_zero<<<4096, T, 0, stream>>>(zbase, zcount);

    // ---- layer 1 ----
    {
        int tiles = (N >> 4) * (H1C1 >> 4);            // 3125 * 10
        int wavesPerBlk = T >> 5;
        k_gemm_wmma_f32<<<(tiles + wavesPerBlk - 1) / wavesPerBlk, T, 0, stream>>>(
            x, W1, h1, N, F_IN, H1C1);
    }
    k_attn_proj<<<((N * HEADS1) + T - 1) / T, T, 0, stream>>>(
        h1, att_src1, att_dst1, as1, ad1, N, HEADS1, CH1);

    int work1 = Etot * HEADS1;
    k_edge_max   <<<(work1 + T - 1) / T, T, 0, stream>>>(ei, E, Etot, HEADS1, as1, ad1, m1);
    k_edge_expsum<<<(work1 + T - 1) / T, T, 0, stream>>>(ei, E, Etot, HEADS1, as1, ad1, m1, den1);
    k_edge_scatter<<<(work1 + T - 1) / T, T, 0, stream>>>(ei, E, Etot, HEADS1, CH1,
                                                          as1, ad1, m1, den1, h1, agg1);
    k_bias_relu<<<((N * H1C1) + T - 1) / T, T, 0, stream>>>(agg1, b1, N, H1C1);

    // ---- layer 2 ----
    {
        int tiles = (N >> 4) * (NCLS >> 4);            // 3125 * 1
        int wavesPerBlk = T >> 5;
        k_gemm_wmma_f32<<<(tiles + wavesPerBlk - 1) / wavesPerBlk, T, 0, stream>>>(
            agg1, W2, h2, N, H1C1, NCLS);
    }
    k_attn_proj<<<(N + T - 1) / T, T, 0, stream>>>(
        h2, att_src2, att_dst2, as2, ad2, N, 1, NCLS);

    k_edge_max   <<<(Etot + T - 1) / T, T, 0, stream>>>(ei, E, Etot, 1, as2, ad2, m2);
    k_edge_expsum<<<(Etot + T - 1) / T, T, 0, stream>>>(ei, E, Etot, 1, as2, ad2, m2, den2);
    k_edge_scatter<<<(Etot + T - 1) / T, T, 0, stream>>>(ei, E, Etot, 1, NCLS,
                                                         as2, ad2, m2, den2, h2, agg2);
    k_bias_out<<<((N * NCLS) + T - 1) / T, T, 0, stream>>>(agg2, b2, out, N, NCLS);
}